// PagedMQA_5093831213086
// MI455X (gfx1250) — compile-verified
//
#include <hip/hip_runtime.h>
#include <hip/hip_bf16.h>

typedef __attribute__((ext_vector_type(2))) float v2f;
typedef __attribute__((ext_vector_type(8))) float v8f;

#define EMBED 2048
#define HD 128
#define NHEADS 16
#define STOT 4096
#define NBATCH 64
#define NSPLIT 32           // waves per batch (8 WGs x 4 waves)
#define TILES_PER_WAVE 8    // 8 tiles * 16 slots = 128 slots per wave

static __device__ __forceinline__ v8f wmma_f32(v2f a, v2f b, v8f c) {
    return __builtin_amdgcn_wmma_f32_16x16x4_f32(false, a, false, b, (short)0, c, false, false);
}

// ---------------------------------------------------------------------------
// GEMM: C[64,N] = A[64,K] @ W[K,N] + bias[N]   (f32 WMMA 16x16x4)
// One wave per 16-wide column tile, covering all 4 row tiles (M=64).
// grid.x = N/64, block = 128 (4 waves).
// ---------------------------------------------------------------------------
__global__ __launch_bounds__(128) void gemm64_kernel(
    const float* __restrict__ A, const float* __restrict__ W,
    const float* __restrict__ bias, float* __restrict__ C, int K, int N)
{
    const int lane = threadIdx.x & 31;
    const int w    = threadIdx.x >> 5;
    const int n    = lane & 15;
    const int hi   = lane >> 4;
    const int col  = (blockIdx.x * 4 + w) * 16 + n;

    v8f acc0 = {}, acc1 = {}, acc2 = {}, acc3 = {};

    #pragma unroll 2
    for (int kc = 0; kc < K / 4; ++kc) {
        const int k0 = 4 * kc + 2 * hi;
        v2f b;
        b.x = W[(size_t)k0 * N + col];
        b.y = W[(size_t)(k0 + 1) * N + col];
        v2f a0 = *(const v2f*)(A + (size_t)(0 * 16 + n) * K + k0);
        v2f a1 = *(const v2f*)(A + (size_t)(1 * 16 + n) * K + k0);
        v2f a2 = *(const v2f*)(A + (size_t)(2 * 16 + n) * K + k0);
        v2f a3 = *(const v2f*)(A + (size_t)(3 * 16 + n) * K + k0);
        acc0 = wmma_f32(a0, b, acc0);
        acc1 = wmma_f32(a1, b, acc1);
        acc2 = wmma_f32(a2, b, acc2);
        acc3 = wmma_f32(a3, b, acc3);
    }

    const float bb = bias[col];
    #pragma unroll
    for (int j = 0; j < 8; ++j) {
        const int row = j + 8 * hi;
        C[(size_t)(0 * 16 + row) * N + col] = acc0[j] + bb;
        C[(size_t)(1 * 16 + row) * N + col] = acc1[j] + bb;
        C[(size_t)(2 * 16 + row) * N + col] = acc2[j] + bb;
        C[(size_t)(3 * 16 + row) * N + col] = acc3[j] + bb;
    }
}

// ---------------------------------------------------------------------------
// Flash-decoding MQA attention, split-K over cache slots.
// grid.x = NBATCH*8, block = 128 (4 waves). Wave = 128 slots (8 tiles of 16).
// Emits per-split partials (running max m, denom l, unnormalized O 16x128).
// ---------------------------------------------------------------------------
__global__ __launch_bounds__(128) void attn_kernel(
    const float* __restrict__ q,      // [64, 2048]  (16 heads x 128)
    const float* __restrict__ kcache, // [64, 4096, 128]
    const float* __restrict__ vcache, // [64, 4096, 128]
    const float* __restrict__ knew,   // [64, 128]
    const float* __restrict__ vnew,   // [64, 128]
    const int*   __restrict__ pos,    // [64]
    float* __restrict__ ml,           // [64, 32, 16, 2]
    float* __restrict__ opart)        // [64, 32, 16, 128]
{
    __shared__ float lds_q[EMBED];
    __shared__ float lds_p[4 * 256];

    const int tid   = threadIdx.x;
    const int w     = tid >> 5;
    const int lane  = tid & 31;
    const int n     = lane & 15;   // head row for A / slot or dim col for B,C
    const int hi    = lane >> 4;
    const int b     = blockIdx.x >> 3;
    const int split = (blockIdx.x & 7) * 4 + w;    // 0..31
    const int pslot = pos[b];
    const size_t kvb = (size_t)b * STOT * HD;

    // stage Q[b] (16x128) into LDS, coalesced
    const float* qg = q + (size_t)b * EMBED;
    for (int i = tid; i < EMBED; i += 128) lds_q[i] = qg[i];
    __syncthreads();

    const float scale = 0.08838834764831845f;  // 1/sqrt(128)

    float m_run[8], l_run[8];
    v8f O[8];
    #pragma unroll
    for (int j = 0; j < 8; ++j) { m_run[j] = -3.0e38f; l_run[j] = 0.0f; }
    #pragma unroll
    for (int f = 0; f < 8; ++f) O[f] = v8f{};

    float* sp = lds_p + w * 256;

    #pragma unroll 1
    for (int t = 0; t < TILES_PER_WAVE; ++t) {
        const int slot_base = split * 128 + t * 16;

        // ---- S = Q @ K^T for a 16-head x 16-slot tile -------------------
        const int ks = slot_base + n;   // this lane's slot column
        const float* kr = (ks == pslot) ? (knew + (size_t)b * HD)
                                        : (kcache + kvb + (size_t)ks * HD);
        v8f S = {};
        #pragma unroll
        for (int kk = 0; kk < 32; ++kk) {
            v2f a = *(const v2f*)(lds_q + n * HD + 4 * kk + 2 * hi);
            v2f kb = *(const v2f*)(kr + 4 * kk + 2 * hi);
            S = wmma_f32(a, kb, S);
        }

        // ---- online softmax (rows j+8*hi live in 16-lane halves) --------
        float pj[8];
        #pragma unroll
        for (int j = 0; j < 8; ++j) {
            float s = S[j] * scale;
            float tm = s;
            tm = fmaxf(tm, __shfl_xor(tm, 1, 32));
            tm = fmaxf(tm, __shfl_xor(tm, 2, 32));
            tm = fmaxf(tm, __shfl_xor(tm, 4, 32));
            tm = fmaxf(tm, __shfl_xor(tm, 8, 32));
            const float mnew  = fmaxf(m_run[j], tm);
            const float alpha = __expf(m_run[j] - mnew);
            const float p     = __expf(s - mnew);
            float ts = p;
            ts += __shfl_xor(ts, 1, 32);
            ts += __shfl_xor(ts, 2, 32);
            ts += __shfl_xor(ts, 4, 32);
            ts += __shfl_xor(ts, 8, 32);
            l_run[j] = l_run[j] * alpha + ts;
            m_run[j] = mnew;
            pj[j] = p;
            #pragma unroll
            for (int f = 0; f < 8; ++f) O[f][j] *= alpha;
        }

        // ---- C-layout -> A-layout for P via per-wave LDS staging --------
        #pragma unroll
        for (int j = 0; j < 8; ++j) sp[(j + 8 * hi) * 16 + n] = pj[j];
        // same-wave DS ops are in-order; no barrier needed

        // ---- O += P @ V  (K = 16 slots, N = 128 dims in 8 fragments) ----
        #pragma unroll
        for (int c4 = 0; c4 < 4; ++c4) {
            v2f a = *(const v2f*)(sp + n * 16 + 4 * c4 + 2 * hi);
            const int s0 = slot_base + 4 * c4 + 2 * hi;
            const int s1 = s0 + 1;
            const float* vr0 = (s0 == pslot) ? (vnew + (size_t)b * HD)
                                             : (vcache + kvb + (size_t)s0 * HD);
            const float* vr1 = (s1 == pslot) ? (vnew + (size_t)b * HD)
                                             : (vcache + kvb + (size_t)s1 * HD);
            #pragma unroll
            for (int f = 0; f < 8; ++f) {
                v2f vb;
                vb.x = vr0[16 * f + n];
                vb.y = vr1[16 * f + n];
                O[f] = wmma_f32(a, vb, O[f]);
            }
        }
    }

    // ---- emit partials --------------------------------------------------
    float* op = opart + ((size_t)b * NSPLIT + split) * NHEADS * HD;
    #pragma unroll
    for (int f = 0; f < 8; ++f)
        #pragma unroll
        for (int j = 0; j < 8; ++j)
            op[(size_t)(j + 8 * hi) * HD + 16 * f + n] = O[f][j];

    if (n == 0) {  // lane 0 (rows 0..7) and lane 16 (rows 8..15)
        float* mlp = ml + ((size_t)b * NSPLIT + split) * NHEADS * 2;
        #pragma unroll
        for (int j = 0; j < 8; ++j) {
            mlp[(j + 8 * hi) * 2 + 0] = m_run[j];
            mlp[(j + 8 * hi) * 2 + 1] = l_run[j];
        }
    }
}

// ---------------------------------------------------------------------------
// Combine 32 split partials per (b,h): attn[b, h*128+d]
// grid.x = 64*16, block = 128 (one thread per dim).
// ---------------------------------------------------------------------------
__global__ __launch_bounds__(128) void reduce_kernel(
    const float* __restrict__ ml, const float* __restrict__ opart,
    float* __restrict__ attn)
{
    const int b = blockIdx.x >> 4;
    const int h = blockIdx.x & 15;
    const int d = threadIdx.x;
    const float* mlb = ml + (size_t)b * NSPLIT * NHEADS * 2;
    const float* ob  = opart + (size_t)b * NSPLIT * NHEADS * HD;

    float M = -3.0e38f;
    for (int s = 0; s < NSPLIT; ++s) M = fmaxf(M, mlb[(s * NHEADS + h) * 2]);

    float L = 0.0f, acc = 0.0f;
    for (int s = 0; s < NSPLIT; ++s) {
        const float ms = mlb[(s * NHEADS + h) * 2 + 0];
        const float ls = mlb[(s * NHEADS + h) * 2 + 1];
        const float wt = __expf(ms - M);
        L   += ls * wt;
        acc += wt * ob[((size_t)s * NHEADS + h) * HD + d];
    }
    attn[(size_t)b * EMBED + h * HD + d] = acc / L;
}

// ---------------------------------------------------------------------------
extern "C" void kernel_launch(void* const* d_in, const int* in_sizes, int n_in,
                              void* d_out, int out_size, void* d_ws, size_t ws_size,
                              hipStream_t stream) {
    const float* x       = (const float*)d_in[0];
    const float* k_cache = (const float*)d_in[1];
    const float* v_cache = (const float*)d_in[2];
    const int*   pos     = (const int*)  d_in[3];
    const float* Wq      = (const float*)d_in[4];
    const float* bq      = (const float*)d_in[5];
    const float* Wk      = (const float*)d_in[6];
    const float* bk      = (const float*)d_in[7];
    const float* Wv      = (const float*)d_in[8];
    const float* bv      = (const float*)d_in[9];
    const float* Wo      = (const float*)d_in[10];
    const float* bo      = (const float*)d_in[11];

    float* ws    = (float*)d_ws;
    float* q     = ws;                       // 64*2048
    float* knew  = q + NBATCH * EMBED;       // 64*128
    float* vnew  = knew + NBATCH * HD;       // 64*128
    float* attn  = vnew + NBATCH * HD;       // 64*2048
    float* ml    = attn + NBATCH * EMBED;    // 64*32*16*2
    float* opart = ml + NBATCH * NSPLIT * NHEADS * 2;  // 64*32*16*128

    // QKV projections
    gemm64_kernel<<<EMBED / 64, 128, 0, stream>>>(x, Wq, bq, q,    EMBED, EMBED);
    gemm64_kernel<<<HD / 64,    128, 0, stream>>>(x, Wk, bk, knew, EMBED, HD);
    gemm64_kernel<<<HD / 64,    128, 0, stream>>>(x, Wv, bv, vnew, EMBED, HD);

    // flash-decoding attention over paged cache (+ scattered new token)
    attn_kernel<<<NBATCH * 8, 128, 0, stream>>>(q, k_cache, v_cache, knew, vnew,
                                                pos, ml, opart);
    reduce_kernel<<<NBATCH * NHEADS, 128, 0, stream>>>(ml, opart, attn);

    // output projection
    gemm64_kernel<<<EMBED / 64, 128, 0, stream>>>(attn, Wo, bo, (float*)d_out,
                                                  EMBED, EMBED);
}